// HGNNModel_35880156791576
// MI455X (gfx1250) — compile-verified
//
#include <hip/hip_runtime.h>
#include <hip/hip_bf16.h>

#define DIM      128
#define LN_EPS   1e-5f
#define LEAKY_K  0.5f

#define WAVES_PER_BLOCK 8
#define NBUF  8      // LDS row slots per wave (power of two)
#define DEPTH 6      // async row fetches in flight (< NBUF)

// ---------------------------------------------------------------------------
// s_wait_asynccnt: builtin if the toolchain has it, else raw SOPP.
// Both variants act as a compiler memory barrier so LDS reads cannot be
// hoisted above the wait.
// ---------------------------------------------------------------------------
#if defined(__has_builtin)
#if __has_builtin(__builtin_amdgcn_s_wait_asynccnt)
#define WAIT_ASYNC(n) do { __builtin_amdgcn_s_wait_asynccnt(n); \
                           asm volatile("" ::: "memory"); } while (0)
#endif
#endif
#ifndef WAIT_ASYNC
#define WAIT_ASYNC(n) asm volatile("s_wait_asynccnt %0" :: "i"(n) : "memory")
#endif

// Per ISA 10.2: flat addresses in the shared aperture have addr[31:0] == LDS
// byte offset (aperture base low 32 bits are zero), so truncating a generic
// pointer to a __shared__ object yields the LDS address the async-load VDST
// operand wants.
static __device__ __forceinline__ unsigned lds_addr32(const void* p) {
    return (unsigned)(unsigned long long)p;
}

// Issue one per-lane 16B async copy global->LDS (ASYNCcnt tracked).
// GVS addressing: mem = SGPR64(base) + VGPR32(voff).
static __device__ __forceinline__ void async_row_fetch(const float* base,
                                                       unsigned voff,
                                                       unsigned ldsoff) {
    asm volatile("global_load_async_to_lds_b128 %0, %1, %2"
                 :: "v"(ldsoff), "v"(voff), "s"(base)
                 : "memory");
}

// ---------------------------------------------------------------------------
// Zero-fill (graph-capture-safe)
// ---------------------------------------------------------------------------
__global__ void zero_f4_kernel(float4* __restrict__ p, int n4) {
    int i = blockIdx.x * blockDim.x + threadIdx.x;
    const int stride = gridDim.x * blockDim.x;
    const float4 z = make_float4(0.f, 0.f, 0.f, 0.f);
    for (; i < n4; i += stride) p[i] = z;
}

// ---------------------------------------------------------------------------
// COO SpMM: out[sidx[e], :] += vals[e] * x[gidx[e], :]
// One wave32 per chunk of 32 nonzeros. The 512B row gathers are staged
// through LDS with GLOBAL_LOAD_ASYNC_TO_LDS_B128 in a depth-6 software
// pipeline (zero VGPR cost for in-flight rows -> high occupancy to hide the
// L2 atomic latency), consumed with ds_load_b128, scattered with 4
// non-returning global_atomic_add_f32 per lane. Whole working set is
// L2-resident on MI455X (192MB L2 vs ~141MB hot data).
//
// ASYNCcnt discipline (loads complete in order):
//   main loop: rows 0..j+DEPTH issued  =>  "asynccnt<=DEPTH" proves row j done
//   tail:      no more issues          =>  one "asynccnt<=0" drains all
// ---------------------------------------------------------------------------
__global__ __launch_bounds__(WAVES_PER_BLOCK * 32) void spmm_coo_kernel(
        const float* __restrict__ vals,
        const int*   __restrict__ gidx,
        const int*   __restrict__ sidx,
        const float* __restrict__ x,
        float*       __restrict__ out,
        int nnz) {
    __shared__ float4 stage[WAVES_PER_BLOCK][NBUF][32];   // 8*8*512B = 32KB

    const int lane   = threadIdx.x & 31;
    const int wib    = threadIdx.x >> 5;
    const int wave   = (blockIdx.x * blockDim.x + threadIdx.x) >> 5;
    const int nwaves = (gridDim.x * blockDim.x) >> 5;

    // per-lane LDS byte address of this wave's slot 0, lane's float4
    const unsigned lds0 = lds_addr32(&stage[wib][0][lane]);

    for (int base = wave * 32; base < nnz; base += nwaves * 32) {
        const int e = base + lane;
        float v = 0.f; int g = 0, s = 0;
        if (e < nnz) { v = vals[e]; g = gidx[e]; s = sidx[e]; }
        const int rem = nnz - base;
        const int cnt = rem < 32 ? rem : 32;

        // ---- prologue: fill the async pipeline ----
        for (int p = 0; p < DEPTH && p < cnt; ++p) {
            const int gp = __shfl(g, p, 32);
            const unsigned voff = ((unsigned)gp * DIM + (unsigned)lane * 4u) * 4u;
            async_row_fetch(x, voff, lds0 + (unsigned)(p & (NBUF - 1)) * (DIM * 4u));
        }

        // ---- steady state: exactly DEPTH loads newer than row j ----
        int j = 0;
        for (; j + DEPTH < cnt; ++j) {
            const int jn = j + DEPTH;
            const int gn = __shfl(g, jn, 32);
            const unsigned voff = ((unsigned)gn * DIM + (unsigned)lane * 4u) * 4u;
            async_row_fetch(x, voff, lds0 + (unsigned)(jn & (NBUF - 1)) * (DIM * 4u));

            WAIT_ASYNC(DEPTH);                    // oldest in-flight = row j
            const float4 xr = stage[wib][j & (NBUF - 1)][lane];   // ds_load_b128

            const float vj = __shfl(v, j, 32);
            const int   sj = __shfl(s, j, 32);
            float* o = out + (size_t)sj * DIM + lane * 4;
            unsafeAtomicAdd(o + 0, vj * xr.x);
            unsafeAtomicAdd(o + 1, vj * xr.y);
            unsafeAtomicAdd(o + 2, vj * xr.z);
            unsafeAtomicAdd(o + 3, vj * xr.w);
        }

        // ---- tail: no further issues; drain once, then consume from LDS ----
        WAIT_ASYNC(0);
        for (; j < cnt; ++j) {
            const float4 xr = stage[wib][j & (NBUF - 1)][lane];
            const float vj = __shfl(v, j, 32);
            const int   sj = __shfl(s, j, 32);
            float* o = out + (size_t)sj * DIM + lane * 4;
            unsafeAtomicAdd(o + 0, vj * xr.x);
            unsafeAtomicAdd(o + 1, vj * xr.y);
            unsafeAtomicAdd(o + 2, vj * xr.z);
            unsafeAtomicAdd(o + 3, vj * xr.w);
        }
        // ASYNCcnt == 0 here; LDS slots free for the next chunk
    }
}

// ---------------------------------------------------------------------------
// x = LayerNorm(leaky?(h)) * gamma + beta + res     (in place over h)
// One wave32 per row; float4 per lane; __shfl_xor butterfly reductions.
// ---------------------------------------------------------------------------
__global__ __launch_bounds__(256) void ln_res_kernel(
        float*       __restrict__ h,
        const float* __restrict__ res,
        const float* __restrict__ gamma,
        const float* __restrict__ beta,
        int n, int apply_leaky) {
    const int lane = threadIdx.x & 31;
    const int row  = (blockIdx.x * blockDim.x + threadIdx.x) >> 5;
    if (row >= n) return;

    float4 hv = ((const float4*)(h + (size_t)row * DIM))[lane];
    if (apply_leaky) {
        hv.x = hv.x > 0.f ? hv.x : LEAKY_K * hv.x;
        hv.y = hv.y > 0.f ? hv.y : LEAKY_K * hv.y;
        hv.z = hv.z > 0.f ? hv.z : LEAKY_K * hv.z;
        hv.w = hv.w > 0.f ? hv.w : LEAKY_K * hv.w;
    }

    float sum = hv.x + hv.y + hv.z + hv.w;
#pragma unroll
    for (int m = 16; m > 0; m >>= 1) sum += __shfl_xor(sum, m, 32);
    const float mu = sum * (1.0f / (float)DIM);

    const float dx = hv.x - mu, dy = hv.y - mu, dz = hv.z - mu, dw = hv.w - mu;
    float sq = dx * dx + dy * dy + dz * dz + dw * dw;
#pragma unroll
    for (int m = 16; m > 0; m >>= 1) sq += __shfl_xor(sq, m, 32);
    const float rstd = rsqrtf(sq * (1.0f / (float)DIM) + LN_EPS);

    const float4 gv = ((const float4*)gamma)[lane];
    const float4 bv = ((const float4*)beta)[lane];
    const float4 rv = ((const float4*)(res + (size_t)row * DIM))[lane];

    float4 o;
    o.x = dx * rstd * gv.x + bv.x + rv.x;
    o.y = dy * rstd * gv.y + bv.y + rv.y;
    o.z = dz * rstd * gv.z + bv.z + rv.z;
    o.w = dw * rstd * gv.w + bv.w + rv.w;
    ((float4*)(h + (size_t)row * DIM))[lane] = o;
}

// ---------------------------------------------------------------------------
// Driver: 2 layers of  x = LN(leaky?(A * (A^T * x))) + ego
//   spmm_t: gather rows, scatter cols ;  spmm: gather cols, scatter rows
// Scratch: one N*DIM f32 buffer in d_ws; d_out doubles as the x/h buffer.
// ---------------------------------------------------------------------------
extern "C" void kernel_launch(void* const* d_in, const int* in_sizes, int n_in,
                              void* d_out, int out_size, void* d_ws, size_t ws_size,
                              hipStream_t stream) {
    const float* ego   = (const float*)d_in[0];
    const float* vals  = (const float*)d_in[1];
    const float* gamma = (const float*)d_in[2];
    const float* beta  = (const float*)d_in[3];
    const int*   rows  = (const int*)d_in[4];
    const int*   cols  = (const int*)d_in[5];

    const int n   = in_sizes[0] / DIM;   // 100000 nodes
    const int nnz = in_sizes[1];         // 3.2M

    float* t  = (float*)d_ws;            // n*DIM scratch (51.2 MB)
    float* xo = (float*)d_out;           // n*DIM: h / x / final output

    const int n4       = n * (DIM / 4);
    const int zblocks  = (n4 + 255) / 256;
    const int chunks   = (nnz + 31) / 32;
    const int sblocks  = (chunks + WAVES_PER_BLOCK - 1) / WAVES_PER_BLOCK;
    const int lnblocks = (n + 7) / 8;

    // ---- Layer 0 ----
    zero_f4_kernel<<<zblocks, 256, 0, stream>>>((float4*)t, n4);
    spmm_coo_kernel<<<sblocks, 256, 0, stream>>>(vals, rows, cols, ego, t, nnz); // t = A^T ego
    zero_f4_kernel<<<zblocks, 256, 0, stream>>>((float4*)xo, n4);
    spmm_coo_kernel<<<sblocks, 256, 0, stream>>>(vals, cols, rows, t, xo, nnz);  // h = A t
    ln_res_kernel<<<lnblocks, 256, 0, stream>>>(xo, ego, gamma, beta, n, 1);     // x1 (leaky)

    // ---- Layer 1 ----
    zero_f4_kernel<<<zblocks, 256, 0, stream>>>((float4*)t, n4);
    spmm_coo_kernel<<<sblocks, 256, 0, stream>>>(vals, rows, cols, xo, t, nnz);  // t = A^T x1
    zero_f4_kernel<<<zblocks, 256, 0, stream>>>((float4*)xo, n4);                // x1 consumed
    spmm_coo_kernel<<<sblocks, 256, 0, stream>>>(vals, cols, rows, t, xo, nnz);  // h = A t
    ln_res_kernel<<<lnblocks, 256, 0, stream>>>(xo, ego, gamma + DIM, beta + DIM, n, 0);
}